// AutoEncoder_21869973471933
// MI455X (gfx1250) — compile-verified
//
#include <hip/hip_runtime.h>
#include <stdint.h>

// Problem constants (match the reference).
#define NF 500000
#define NV 250000
#define DD 192
#define NC (3 * NF)          // 1.5M face-corners
#define N4 (NV * (DD / 4))   // output in float4 units = 12M

// ---------------------------------------------------------------------------
// Scatter-add. One wave32 per corner (grid-stride, persistent).
// Each corner's 768B feature chunk is staged into LDS via CDNA5 async
// global->LDS DMA (3x global_load_async_to_lds_b64 = 3x256B), double-buffered
// (ASYNCcnt) so the next corner streams in while the current one is being
// atomically added with no-return global_atomic_add_f32 at device scope.
// ---------------------------------------------------------------------------
__global__ __launch_bounds__(256, 4) void scatter_kernel(const float* __restrict__ ff,
                                                         const int* __restrict__ faces,
                                                         float* __restrict__ out,
                                                         unsigned* __restrict__ cnt) {
    // 8 waves/block * 2 buffers * 192 floats = 12,288 B of LDS.
    __shared__ __align__(16) float smem[8][2][DD];

    const int lane = threadIdx.x & 31;
    const int w    = threadIdx.x >> 5;
    // Wave-uniform ids: readfirstlane so the compiler scalarizes index math
    // and the faces[] fetch becomes an s_load.
    const int gw = __builtin_amdgcn_readfirstlane(blockIdx.x * 8 + w);
    const int nw = gridDim.x * 8;

    int corner = gw;
    if (corner >= NC) return;                   // uniform: EXEC stays all-ones

    // LDS byte addresses of this wave's two staging buffers (per-lane slot).
    const unsigned lds0 = (unsigned)(uintptr_t)(&smem[w][0][0]) + (unsigned)(lane * 8);
    const unsigned lds1 = (unsigned)(uintptr_t)(&smem[w][1][0]) + (unsigned)(lane * 8);

    // Prime the pipeline: async-copy corner's 768B into buffer 0.
    {
        unsigned voff = (unsigned)corner * 768u + (unsigned)(lane * 8);
        asm volatile(
            "global_load_async_to_lds_b64 %0, %1, %2 offset:0\n\t"
            "global_load_async_to_lds_b64 %0, %1, %2 offset:256\n\t"
            "global_load_async_to_lds_b64 %0, %1, %2 offset:512"
            :: "v"(lds0), "v"(voff), "s"(ff) : "memory");
    }

    int buf = 0;
    while (corner < NC) {
        // Issue the (scalar) vertex-index load BEFORE the asm memory-clobber
        // barriers so it overlaps the async wait.
        const int vtx = faces[corner];

        const int next = corner + nw;
        if (next < NC) {
            // Prefetch next corner into the other buffer, then wait for the
            // current buffer's 3 transfers (leave the new 3 in flight).
            unsigned voff = (unsigned)next * 768u + (unsigned)(lane * 8);
            unsigned ldsn = buf ? lds0 : lds1;
            asm volatile(
                "global_load_async_to_lds_b64 %0, %1, %2 offset:0\n\t"
                "global_load_async_to_lds_b64 %0, %1, %2 offset:256\n\t"
                "global_load_async_to_lds_b64 %0, %1, %2 offset:512"
                :: "v"(ldsn), "v"(voff), "s"(ff) : "memory");
            asm volatile("s_wait_asynccnt 0x3" ::: "memory");
        } else {
            asm volatile("s_wait_asynccnt 0x0" ::: "memory");
        }

        // Consume: lane-strided reads (bank-conflict-free), then 6 coalesced
        // no-return f32 atomics per lane into the destination vertex row.
        const float* sp = &smem[w][buf][lane];
        float a0 = sp[0 * 32], a1 = sp[1 * 32], a2 = sp[2 * 32];
        float a3 = sp[3 * 32], a4 = sp[4 * 32], a5 = sp[5 * 32];

        const unsigned dstoff = (unsigned)vtx * (DD * 4u) + (unsigned)(lane * 4);
        asm volatile(
            "global_atomic_add_f32 %6, %0, %7 scope:SCOPE_DEV\n\t"
            "global_atomic_add_f32 %6, %1, %7 offset:128 scope:SCOPE_DEV\n\t"
            "global_atomic_add_f32 %6, %2, %7 offset:256 scope:SCOPE_DEV\n\t"
            "global_atomic_add_f32 %6, %3, %7 offset:384 scope:SCOPE_DEV\n\t"
            "global_atomic_add_f32 %6, %4, %7 offset:512 scope:SCOPE_DEV\n\t"
            "global_atomic_add_f32 %6, %5, %7 offset:640 scope:SCOPE_DEV"
            :: "v"(a0), "v"(a1), "v"(a2), "v"(a3), "v"(a4), "v"(a5),
               "v"(dstoff), "s"(out)
            : "memory");

        if (lane == 0) {
            __hip_atomic_fetch_add(&cnt[vtx], 1u, __ATOMIC_RELAXED,
                                   __HIP_MEMORY_SCOPE_AGENT);
        }

        buf ^= 1;
        corner = next;
    }
}

// ---------------------------------------------------------------------------
// Zero the output accumulator and the per-vertex hit counters.
// ---------------------------------------------------------------------------
__global__ __launch_bounds__(256) void init_zero_kernel(float4* __restrict__ out4,
                                                        unsigned* __restrict__ cnt) {
    const int stride = gridDim.x * blockDim.x;
    int i = blockIdx.x * blockDim.x + threadIdx.x;
    for (int k = i; k < N4; k += stride) out4[k] = make_float4(0.f, 0.f, 0.f, 0.f);
    for (int k = i; k < NV; k += stride) cnt[k] = 0u;
}

// ---------------------------------------------------------------------------
// Divide each vertex row by max(count, 1). float4-vectorized; 192 % 4 == 0 so
// every float4 belongs to exactly one vertex.
// ---------------------------------------------------------------------------
__global__ __launch_bounds__(256) void finalize_kernel(float4* __restrict__ out4,
                                                       const unsigned* __restrict__ cnt) {
    const int stride = gridDim.x * blockDim.x;
    for (int k = blockIdx.x * blockDim.x + threadIdx.x; k < N4; k += stride) {
        const unsigned c = cnt[k / (DD / 4)];
        const float inv = 1.0f / fmaxf((float)c, 1.0f);
        float4 v = out4[k];
        v.x *= inv; v.y *= inv; v.z *= inv; v.w *= inv;
        out4[k] = v;
    }
}

// ---------------------------------------------------------------------------
// Host-side launcher.
// ---------------------------------------------------------------------------
extern "C" void kernel_launch(void* const* d_in, const int* in_sizes, int n_in,
                              void* d_out, int out_size, void* d_ws, size_t ws_size,
                              hipStream_t stream) {
    const float* ff    = (const float*)d_in[0];   // (F, 576) face features
    const int*   faces = (const int*)d_in[1];     // (F, 3) vertex ids
    float*       out   = (float*)d_out;           // (V, 192)
    unsigned*    cnt   = (unsigned*)d_ws;         // V u32 hit counters (scratch)

    (void)in_sizes; (void)n_in; (void)out_size; (void)ws_size;

    init_zero_kernel<<<2048, 256, 0, stream>>>((float4*)out, cnt);
    scatter_kernel<<<8192, 256, 0, stream>>>(ff, faces, out, cnt);
    finalize_kernel<<<2048, 256, 0, stream>>>((float4*)out, cnt);
}